// TopDownRNNG_39204461478342
// MI455X (gfx1250) — compile-verified
//
#include <hip/hip_runtime.h>
#include <hip/hip_bf16.h>
#include <math.h>

#define BB 16384
#define LL 50
#define WW 128
#define HH 128

// Padded LDS row stride (floats) for the 16x256 A tile:
// 260 mod 64 = 4  ->  bank(row r, col k) = (4r + k) mod 64, conflict-free for
// the 16 rows read by each half-wave, and the K+2 half-wave hits the odd set.
#define ASTRIDE 260

typedef __attribute__((ext_vector_type(2))) float v2f;
typedef __attribute__((ext_vector_type(8))) float v8f;

__device__ __forceinline__ float sigf(float x) { return 1.0f / (1.0f + __expf(-x)); }

// Async-copy one float4 from global to LDS (GLOBAL_LOAD_ASYNC_TO_LDS_B128,
// GV mode, tracked by ASYNCcnt). ldsoff = low 32 bits of generic LDS pointer.
__device__ __forceinline__ void async_g2l_b128(const float* gsrc, float* ldst) {
    const unsigned int ldsoff = (unsigned int)(unsigned long long)ldst;
    asm volatile("global_load_async_to_lds_b128 %0, %1, off"
                 :: "v"(ldsoff), "v"(gsrc) : "memory");
}
__device__ __forceinline__ void async_wait_all() {
    asm volatile("s_wait_asynccnt 0x0" ::: "memory");
}

// ---------------------------------------------------------------------------
// Kernel 1: u[b, n] = sum_d rhs[b,d] * V_W[n,d]
//   rhs[b] = [nt_emb_W[nt_id[b]] (128) ; stack_state[b] (128)]   K = 256
//   M = B (16 rows/block), N = 128 (8 waves x 16 cols), WMMA f32 16x16x4.
//   A tile staged via async global->LDS copies, shared by all 8 waves.
// ---------------------------------------------------------------------------
__global__ __launch_bounds__(256) void k_uproj(
    const float* __restrict__ stack_state,
    const float* __restrict__ V_W,        // (128, 256) row-major
    const float* __restrict__ nt_emb_W,   // (100, 128)
    const int*   __restrict__ nt_id,
    float*       __restrict__ u)          // (B, 128)
{
    __shared__ __align__(16) float tileA[16 * ASTRIDE];

    const int tid = threadIdx.x;
    const int b0  = blockIdx.x * 16;

    // Stage concat(nt_emb[nt_id], stack_state) rows b0..b0+15 -> LDS.
    #pragma unroll
    for (int i = 0; i < 2; ++i) {
        const int flat = tid * 2 + i;          // 0..511
        const int row  = flat >> 5;            // 0..15
        const int q    = flat & 31;            // float4 index within 128 cols
        const float* gLow  = nt_emb_W + (size_t)nt_id[b0 + row] * WW + q * 4;
        const float* gHigh = stack_state + (size_t)(b0 + row) * HH + q * 4;
        async_g2l_b128(gLow,  &tileA[row * ASTRIDE + q * 4]);
        async_g2l_b128(gHigh, &tileA[row * ASTRIDE + 128 + q * 4]);
    }
    async_wait_all();
    __syncthreads();

    const int lane = tid & 31;
    const int wv   = tid >> 5;             // 0..7 -> N tile
    const int n0   = wv * 16;
    const int r    = lane & 15;
    const int koff = (lane >> 4) * 2;      // lanes 16..31 carry K+2, K+3

    const float* arow = &tileA[r * ASTRIDE];
    const float* bcol = V_W + (size_t)(n0 + r) * 256;   // V_W[n, :]

    v8f acc = {};
    #pragma unroll 8
    for (int k0 = 0; k0 < 256; k0 += 4) {
        v2f a; a[0] = arow[k0 + koff]; a[1] = arow[k0 + koff + 1];
        v2f b; b[0] = bcol[k0 + koff]; b[1] = bcol[k0 + koff + 1];
        acc = __builtin_amdgcn_wmma_f32_16x16x4_f32(false, a, false, b,
                                                    (short)0, acc, false, false);
    }
    const int rowAdd = (lane < 16) ? 0 : 8;
    #pragma unroll
    for (int t = 0; t < 8; ++t) {
        const int m = b0 + t + rowAdd;
        u[(size_t)m * WW + n0 + r] = acc[t];
    }
}

// ---------------------------------------------------------------------------
// Kernel 2: streaming attention, one wave per batch row, single pass over
// children[b] (online softmax).  logit[l] = children[b,l,:] . u[b,:]
// Writes attn (B,L) and weighted (B,W).  This is the HBM-roofline kernel
// (419 MB of children traffic, read exactly once, float4-coalesced).
// ---------------------------------------------------------------------------
__global__ __launch_bounds__(256) void k_attn(
    const float* __restrict__ children,   // (B, L, W)
    const float* __restrict__ u,          // (B, W)
    const int*   __restrict__ ch_lengths, // (B)
    float*       __restrict__ weighted,   // (B, W)
    float*       __restrict__ attn_out)   // (B, L)
{
    const int lane = threadIdx.x & 31;
    const int b    = blockIdx.x * 8 + (threadIdx.x >> 5);
    const int len  = ch_lengths[b];

    const float4* ch4 = reinterpret_cast<const float4*>(children + (size_t)b * LL * WW);
    const float4  u4  = reinterpret_cast<const float4*>(u + (size_t)b * WW)[lane];

    float  m = -__builtin_inff();
    float  s = 0.0f;
    float4 acc = make_float4(0.f, 0.f, 0.f, 0.f);
    float  lg0 = 0.0f, lg1 = 0.0f;   // distributed logit storage (lane l, lane l+32)

    for (int l = 0; l < len; ++l) {
        const float4 c = ch4[l * 32 + lane];
        if (l + 1 < len) __builtin_prefetch(&ch4[(l + 1) * 32 + lane], 0, 0);

        float p = c.x * u4.x + c.y * u4.y + c.z * u4.z + c.w * u4.w;
        #pragma unroll
        for (int off = 16; off >= 1; off >>= 1)
            p += __shfl_xor(p, off, 32);

        if ((l & 31) == lane) { if (l < 32) lg0 = p; else lg1 = p; }

        const float mn = fmaxf(m, p);
        const float sc = __expf(m - mn);     // 0 on first iteration (m = -inf)
        const float e  = __expf(p - mn);
        s     = s * sc + e;
        acc.x = acc.x * sc + e * c.x;
        acc.y = acc.y * sc + e * c.y;
        acc.z = acc.z * sc + e * c.z;
        acc.w = acc.w * sc + e * c.w;
        m = mn;
    }

    const float inv = 1.0f / s;              // len >= 1 guaranteed
    float4 w4;
    w4.x = acc.x * inv; w4.y = acc.y * inv; w4.z = acc.z * inv; w4.w = acc.w * inv;
    reinterpret_cast<float4*>(weighted + (size_t)b * WW)[lane] = w4;

    // attn: exp(logit - m)/s for l < len, 0 otherwise (softmax of -inf)
    attn_out[(size_t)b * LL + lane] = (lane < len) ? __expf(lg0 - m) * inv : 0.0f;
    const int l1 = 32 + lane;
    if (l1 < LL)
        attn_out[(size_t)b * LL + l1] = (l1 < len) ? __expf(lg1 - m) * inv : 0.0f;
}

// ---------------------------------------------------------------------------
// Kernel 3: g = sigmoid(concat(nt2, weighted) @ gate_W + gate_b)
//           comp = g*nt2 + (1-g)*weighted          (K = 256, N = 128)
// A tile (concat) staged to LDS; epilogue reuses it for nt2/weighted.
// ---------------------------------------------------------------------------
__global__ __launch_bounds__(256) void k_gate(
    const float* __restrict__ nt_emb2_W,  // (100, 128)
    const float* __restrict__ gate_W,     // (256, 128) row-major
    const float* __restrict__ gate_b,     // (128)
    const float* __restrict__ weighted,   // (B, 128)
    const int*   __restrict__ nt_id,
    float*       __restrict__ g_out,      // (B, 128)
    float*       __restrict__ comp_out)   // (B, 128)
{
    __shared__ __align__(16) float tileA[16 * ASTRIDE];

    const int tid = threadIdx.x;
    const int b0  = blockIdx.x * 16;

    #pragma unroll
    for (int i = 0; i < 2; ++i) {
        const int flat = tid * 2 + i;
        const int row  = flat >> 5;
        const int q    = flat & 31;
        const float* gLow  = nt_emb2_W + (size_t)nt_id[b0 + row] * WW + q * 4;
        const float* gHigh = weighted + (size_t)(b0 + row) * WW + q * 4;
        async_g2l_b128(gLow,  &tileA[row * ASTRIDE + q * 4]);
        async_g2l_b128(gHigh, &tileA[row * ASTRIDE + 128 + q * 4]);
    }
    async_wait_all();
    __syncthreads();

    const int lane = tid & 31;
    const int wv   = tid >> 5;
    const int n0   = wv * 16;
    const int r    = lane & 15;
    const int koff = (lane >> 4) * 2;

    const float* arow = &tileA[r * ASTRIDE];

    v8f acc = {};
    #pragma unroll 8
    for (int k0 = 0; k0 < 256; k0 += 4) {
        v2f a; a[0] = arow[k0 + koff]; a[1] = arow[k0 + koff + 1];
        v2f b; b[0] = gate_W[(size_t)(k0 + koff) * WW + n0 + r];
               b[1] = gate_W[(size_t)(k0 + koff + 1) * WW + n0 + r];
        acc = __builtin_amdgcn_wmma_f32_16x16x4_f32(false, a, false, b,
                                                    (short)0, acc, false, false);
    }

    const int rowAdd = (lane < 16) ? 0 : 8;
    const int col    = n0 + r;
    const float bb   = gate_b[col];
    #pragma unroll
    for (int t = 0; t < 8; ++t) {
        const int  tr   = t + rowAdd;              // row within tile
        const int  mrow = b0 + tr;
        const float gv  = sigf(acc[t] + bb);
        const float nt2 = tileA[tr * ASTRIDE + col];        // low half of concat
        const float wvv = tileA[tr * ASTRIDE + 128 + col];  // high half
        g_out[(size_t)mrow * WW + col]    = gv;
        comp_out[(size_t)mrow * WW + col] = gv * nt2 + (1.0f - gv) * wvv;
    }
}

// ---------------------------------------------------------------------------
// Kernel 4/5: one LSTM layer step.
//   all = concat(x, prev_h) @ Wl + bl  (K=256, N=512 as 4 gate blocks of 128)
//   c = sig(f)*prev_c + sig(i)*tanh(g);  h = sig(o)*tanh(c)
// Each wave: 16 rows x 16 j-cols, 4 WMMA accumulators (one per gate).
// concat(x, prev_h) tile staged once per block via async global->LDS.
// ---------------------------------------------------------------------------
__global__ __launch_bounds__(256) void k_lstm(
    const float* __restrict__ x,       // (B, 128)
    const float* __restrict__ ph,      // (B, 128)  prev_h[layer]
    const float* __restrict__ pc,      // (B, 128)  prev_c[layer]
    const float* __restrict__ Wl,      // (256, 512) row-major
    const float* __restrict__ bl,      // (512)
    float*       __restrict__ h_out)   // (B, 128)
{
    __shared__ __align__(16) float tileA[16 * ASTRIDE];

    const int tid = threadIdx.x;
    const int b0  = blockIdx.x * 16;

    #pragma unroll
    for (int i = 0; i < 2; ++i) {
        const int flat = tid * 2 + i;
        const int row  = flat >> 5;
        const int q    = flat & 31;
        const float* gLow  = x  + (size_t)(b0 + row) * WW + q * 4;
        const float* gHigh = ph + (size_t)(b0 + row) * HH + q * 4;
        async_g2l_b128(gLow,  &tileA[row * ASTRIDE + q * 4]);
        async_g2l_b128(gHigh, &tileA[row * ASTRIDE + 128 + q * 4]);
    }
    async_wait_all();
    __syncthreads();

    const int lane = tid & 31;
    const int wv   = tid >> 5;
    const int j0   = wv * 16;
    const int r    = lane & 15;
    const int koff = (lane >> 4) * 2;

    const float* arow = &tileA[r * ASTRIDE];

    v8f ai = {}, af = {}, ao = {}, ag = {};
    #pragma unroll 4
    for (int k0 = 0; k0 < 256; k0 += 4) {
        v2f a; a[0] = arow[k0 + koff]; a[1] = arow[k0 + koff + 1];
        const float* w0 = Wl + (size_t)(k0 + koff) * 512 + j0 + r;
        const float* w1 = w0 + 512;
        v2f b;
        b[0] = w0[0];   b[1] = w1[0];
        ai = __builtin_amdgcn_wmma_f32_16x16x4_f32(false, a, false, b, (short)0, ai, false, false);
        b[0] = w0[128]; b[1] = w1[128];
        af = __builtin_amdgcn_wmma_f32_16x16x4_f32(false, a, false, b, (short)0, af, false, false);
        b[0] = w0[256]; b[1] = w1[256];
        ao = __builtin_amdgcn_wmma_f32_16x16x4_f32(false, a, false, b, (short)0, ao, false, false);
        b[0] = w0[384]; b[1] = w1[384];
        ag = __builtin_amdgcn_wmma_f32_16x16x4_f32(false, a, false, b, (short)0, ag, false, false);
    }

    const int rowAdd = (lane < 16) ? 0 : 8;
    const int col    = j0 + r;
    const float bi_ = bl[col], bf_ = bl[128 + col], bo_ = bl[256 + col], bg_ = bl[384 + col];
    #pragma unroll
    for (int t = 0; t < 8; ++t) {
        const int  mrow = b0 + t + rowAdd;
        const float cv = sigf(af[t] + bf_) * pc[(size_t)mrow * HH + col]
                       + sigf(ai[t] + bi_) * tanhf(ag[t] + bg_);
        h_out[(size_t)mrow * HH + col] = sigf(ao[t] + bo_) * tanhf(cv);
    }
}

// ---------------------------------------------------------------------------
extern "C" void kernel_launch(void* const* d_in, const int* in_sizes, int n_in,
                              void* d_out, int out_size, void* d_ws, size_t ws_size,
                              hipStream_t stream) {
    const float* children    = (const float*)d_in[0];
    const float* stack_state = (const float*)d_in[1];
    const float* prev_h      = (const float*)d_in[2];   // (2, B, H)
    const float* prev_c      = (const float*)d_in[3];   // (2, B, H)
    const float* V_W         = (const float*)d_in[4];
    const float* nt_emb_W    = (const float*)d_in[5];
    const float* nt_emb2_W   = (const float*)d_in[6];
    const float* gate_W      = (const float*)d_in[7];
    const float* gate_b      = (const float*)d_in[8];
    const float* lstm_W      = (const float*)d_in[9];   // (2, 256, 512)
    const float* lstm_b      = (const float*)d_in[10];  // (2, 512)
    const int*   ch_lengths  = (const int*)d_in[11];
    const int*   nt_id       = (const int*)d_in[12];

    float* out      = (float*)d_out;
    float* out_h    = out;                                        // (B, H)
    float* out_comp = out + (size_t)BB * HH;                      // (B, W)
    float* out_attn = out + (size_t)BB * (HH + WW);               // (B, L)
    float* out_g    = out + (size_t)BB * (HH + WW) + (size_t)BB * LL; // (B, W)

    float* ws       = (float*)d_ws;
    float* u        = ws;                         // (B, 128)
    float* weighted = ws + (size_t)BB * WW;       // (B, 128)
    float* h0       = ws + (size_t)2 * BB * WW;   // (B, 128)

    dim3 blk(256);
    k_uproj<<<BB / 16, blk, 0, stream>>>(stack_state, V_W, nt_emb_W, nt_id, u);
    k_attn <<<BB / 8,  blk, 0, stream>>>(children, u, ch_lengths, weighted, out_attn);
    k_gate <<<BB / 16, blk, 0, stream>>>(nt_emb2_W, gate_W, gate_b, weighted, nt_id,
                                         out_g, out_comp);
    k_lstm <<<BB / 16, blk, 0, stream>>>(out_comp, prev_h, prev_c,
                                         lstm_W, lstm_b, h0);
    k_lstm <<<BB / 16, blk, 0, stream>>>(h0, prev_h + (size_t)BB * HH,
                                         prev_c + (size_t)BB * HH,
                                         lstm_W + (size_t)256 * 512, lstm_b + 512, out_h);
}